// TargetPositionPredictor_84232898609302
// MI455X (gfx1250) — compile-verified
//
#include <hip/hip_runtime.h>
#include <math.h>
#include <limits.h>
#include <stdint.h>

// ---------------- problem constants ----------------
#define LMAX        5
#define MUL_IN      128
#define NUM_CH      4
#define NUM_RAD     20
#define RES_BETA    180
#define RES_ALPHA   359
#define IRREPS      36                    // (LMAX+1)^2
#define NG          32                    // graphs
#define NCOL        (RES_BETA * RES_ALPHA)        // 64620 grid points
#define NPAD2       65536                 // columns padded to 4096 n-tiles (power of 2)
#define NTILE_N     (NPAD2 / 16)          // 4096
#define MROWS       (NG * NUM_CH * NUM_RAD)       // 2560
#define NTILE_M     (MROWS / 16)          // 160
#define MGROUPS     (NTILE_M / 8)         // 20  (8 m-tiles per block => 8 waves)
#define CHUNK       64                    // n-tiles handled per block
#define NCHUNKS     (NTILE_N / CHUNK)     // 64
#define STRIP       8                     // n-tiles staged in LDS per barrier
#define BS_STRIDE   37                    // LDS row stride (odd -> bank-conflict-free)
#define COEFF_OUT   (NG * NUM_CH * NUM_RAD * IRREPS)  // 92160 floats
#define LOGITS_PER_G (NUM_RAD * NCOL)     // 1,292,400
#define TOT_LOGITS  (NG * LOGITS_PER_G)   // 41,356,800
#define INV_SQRT_FAN 0.08838834764831845f // 1/sqrt(128)

typedef __attribute__((ext_vector_type(2))) float v2f;
typedef __attribute__((ext_vector_type(8))) float v8f;

// ---- monotonic int key for float atomicMax ----
__device__ __forceinline__ int fkey(float f) {
    int i = __float_as_int(f);
    return (i >= 0) ? i : (i ^ 0x7fffffff);
}
__device__ __forceinline__ float funkey(int k) {
    return __int_as_float((k >= 0) ? k : (k ^ 0x7fffffff));
}

// ================= kernel 1: Gauss-Legendre nodes (f64 Newton) + max-key init =================
__global__ void gl_nodes_kernel(double* __restrict__ nodes, int* __restrict__ maxkey) {
    int i = threadIdx.x;
    if (i < NG) maxkey[i] = INT_MIN;
    if (i < RES_BETA) {
        const int n = RES_BETA;
        double x = cos(M_PI * (i + 0.75) / (n + 0.5));
        for (int it = 0; it < 64; ++it) {
            double p0 = 1.0, p1 = x;
            for (int k = 2; k <= n; ++k) {
                double p2 = ((2.0 * k - 1.0) * x * p1 - (k - 1.0) * p0) / (double)k;
                p0 = p1; p1 = p2;
            }
            double dp = n * (x * p1 - p0) / (x * x - 1.0);
            x -= p1 / dp;
        }
        nodes[(n - 1) - i] = x;   // leggauss returns ascending nodes
    }
}

// ================= kernel 2: build Yt[36][NPAD2] (B-matrix of the big GEMM) =================
__global__ void ygrid_kernel(const double* __restrict__ nodes, float* __restrict__ Yt) {
    int col = blockIdx.x * blockDim.x + threadIdx.x;
    if (col >= NPAD2) return;
    if (col >= NCOL) {                       // zero pad tail columns
        for (int i = 0; i < IRREPS; ++i) Yt[(size_t)i * NPAD2 + col] = 0.0f;
        return;
    }
    int b = col / RES_ALPHA;
    int a = col - b * RES_ALPHA;
    double y = nodes[b];
    // associated Legendre P[l][m], no Condon-Shortley
    double P[LMAX + 1][LMAX + 1];
    double sy = sqrt(fmax(1.0 - y * y, 0.0));
    P[0][0] = 1.0;
    for (int m = 1; m <= LMAX; ++m) P[m][m] = (2.0 * m - 1.0) * sy * P[m - 1][m - 1];
    for (int m = 0; m < LMAX; ++m) P[m + 1][m] = (2.0 * m + 1.0) * y * P[m][m];
    for (int m = 0; m <= LMAX; ++m)
        for (int l = m + 2; l <= LMAX; ++l)
            P[l][m] = ((2.0 * l - 1.0) * y * P[l - 1][m] - (l + m - 1.0) * P[l - 2][m]) / (double)(l - m);
    // Fourier basis via Chebyshev recurrence
    double alpha = 2.0 * M_PI * (double)a / (double)RES_ALPHA;
    double c1 = cos(alpha), s1 = sin(alpha);
    double cosm[LMAX + 1], sinm[LMAX + 1];
    double cm = 1.0, sm = 0.0;
    for (int m = 0; m <= LMAX; ++m) {
        cosm[m] = cm; sinm[m] = sm;
        double cn = cm * c1 - sm * s1;
        sm = sm * c1 + cm * s1;
        cm = cn;
    }
    const double SQRT2 = 1.4142135623730951;
    int idx = 0;
    for (int l = 0; l <= LMAX; ++l) {
        for (int m = -l; m <= l; ++m) {
            int am = m < 0 ? -m : m;
            double ratio = 1.0;                               // (l-am)!/(l+am)!
            for (int k = l - am + 1; k <= l + am; ++k) ratio /= (double)k;
            double K = sqrt((2.0 * l + 1.0) / (4.0 * M_PI) * ratio);
            double v;
            if (m == 0)      v = K * P[l][0];
            else if (m > 0)  v = SQRT2 * K * P[l][am] * cosm[am];
            else             v = SQRT2 * K * P[l][am] * sinm[am];
            Yt[(size_t)idx * NPAD2 + col] = (float)v;
            ++idx;
        }
    }
}

// ================= kernel 3: gated per-l linear -> position_coeffs + reordered A =================
// block = (g, l); out layout [G,C,R,36]; A layout row = g*80 + r*4 + c (channels adjacent)
__global__ void coeffs_kernel(const float* __restrict__ focus,
                              const float* __restrict__ embt,
                              const float* __restrict__ W,
                              const int* __restrict__ species,
                              float* __restrict__ out_coeffs,
                              float* __restrict__ Amat) {
    int g = blockIdx.x / (LMAX + 1);
    int l = blockIdx.x % (LMAX + 1);
    int dim_m = 2 * l + 1;
    int off_in = MUL_IN * l * l;          // sum_{l'<l} 128*(2l'+1) = 128*l^2
    int sp = species[g];

    __shared__ float xe[MUL_IN * (2 * LMAX + 1)];
    for (int idx = threadIdx.x; idx < MUL_IN * dim_m; idx += blockDim.x) {
        int i = idx / dim_m, m = idx - i * dim_m;
        xe[idx] = focus[(size_t)g * (MUL_IN * IRREPS) + off_in + i * dim_m + m]
                * embt[(size_t)sp * (MUL_IN * (LMAX + 1)) + l * MUL_IN + i];
    }
    __syncthreads();

    const float* Wl = W + (size_t)l * MUL_IN * (NUM_RAD * NUM_CH);
    for (int o = threadIdx.x; o < NUM_RAD * NUM_CH; o += blockDim.x) {
        int c = o / NUM_RAD, r = o - c * NUM_RAD;
        for (int m = 0; m < dim_m; ++m) {
            float acc = 0.0f;
            for (int i = 0; i < MUL_IN; ++i)
                acc = fmaf(xe[i * dim_m + m], Wl[i * (NUM_RAD * NUM_CH) + o], acc);
            float v = acc * INV_SQRT_FAN;
            out_coeffs[(((size_t)g * NUM_CH + c) * NUM_RAD + r) * IRREPS + l * l + m] = v;
            Amat[((size_t)g * 80 + r * 4 + c) * IRREPS + l * l + m] = v;
        }
    }
}

// ================= kernel 4: blocked WMMA GEMM + fused logsumexp + graph max =================
// Block = 8 waves covering m-tiles [mgroup*8 .. +7]; walks a 64-n-tile chunk in strips of 8.
// A fragments live in registers for the whole kernel; B strips staged into LDS via
// CDNA5 async global->LDS loads and shared by all 8 waves.
__device__ __forceinline__ float lse4(float x0, float x1, float x2, float x3) {
    float m = fmaxf(fmaxf(x0, x1), fmaxf(x2, x3));
    float s = __expf(x0 - m) + __expf(x1 - m) + __expf(x2 - m) + __expf(x3 - m);
    return m + __logf(s);
}

__global__ void __launch_bounds__(256)
gemm_lse_kernel(const float* __restrict__ Amat,
                const float* __restrict__ Yt,
                float* __restrict__ logits,
                int* __restrict__ maxkey) {
    const int lane = threadIdx.x & 31;
    const int wv   = threadIdx.x >> 5;
    const int mgroup = blockIdx.x % MGROUPS;
    const int nchunk = blockIdx.x / MGROUPS;
    const int m_tile = mgroup * 8 + wv;

    const int half = lane >> 4;           // 0: K pair {0,1}, 1: K pair {2,3}
    const int mn   = lane & 15;           // row (A) / column (B) index within tile
    const int krow = 2 * half;

    // ---- A fragment: 16x36 rows of this wave's m-tile, resident in 18 VGPRs ----
    v2f afrag[9];
    {
        const float* Ar = Amat + ((size_t)(m_tile * 16 + mn)) * IRREPS + krow;
#pragma unroll
        for (int s = 0; s < 9; ++s) { afrag[s].x = Ar[4 * s]; afrag[s].y = Ar[4 * s + 1]; }
    }

    // ---- LDS B strip: [128 cols][36 k], stride 37 (odd -> bank-conflict-free) ----
    __shared__ float Bs[128 * BS_STRIDE];  // 18944 B
    const unsigned bs_base = (unsigned)(uintptr_t)(&Bs[0]);  // low 32 bits = LDS offset
    const unsigned long long yt_base = (unsigned long long)(uintptr_t)Yt;

    // per-thread fill assignment: 4608 floats / 256 threads = 18 each
    const int fk = threadIdx.x >> 7;       // base k contribution (idx>>7 pattern below)
    (void)fk;

    for (int strip = 0; strip < CHUNK / STRIP; ++strip) {
        const int nt0 = nchunk * CHUNK + strip * STRIP;    // first n-tile of strip
        const unsigned long long colbase = (unsigned long long)nt0 * 16u;

        __syncthreads();   // all waves done reading previous strip
        // ---- async global -> LDS staging (CDNA5 GLOBAL_LOAD_ASYNC_TO_LDS_B32) ----
#pragma unroll
        for (int it = 0; it < 18; ++it) {
            int idx = threadIdx.x + it * 256;              // 0 .. 4607
            int k = idx >> 7;                              // 0 .. 35
            int j = idx & 127;                             // 0 .. 127 (col in strip)
            unsigned ldsoff = bs_base + (unsigned)((j * BS_STRIDE + k) * 4);
            unsigned long long ga = yt_base +
                (((unsigned long long)k * NPAD2) + colbase + (unsigned)j) * 4ull;
            asm volatile("global_load_async_to_lds_b32 %0, %1, off"
                         :: "v"(ldsoff), "v"(ga) : "memory");
        }
        asm volatile("s_wait_asynccnt 0x0" ::: "memory");
        __syncthreads();   // strip visible to all waves

        // ---- 8 n-tiles per strip, 9 WMMAs each, A reused from registers ----
        for (int t = 0; t < STRIP; ++t) {
            const float* bcol = &Bs[(t * 16 + mn) * BS_STRIDE + krow];
            v8f acc = {};
#pragma unroll
            for (int s = 0; s < 9; ++s) {
                v2f b;
                b.x = bcol[4 * s];
                b.y = bcol[4 * s + 1];
                acc = __builtin_amdgcn_wmma_f32_16x16x4_f32(false, afrag[s], false, b,
                                                            (short)0, acc, false, false);
            }
            // rows v + 8*half; 4 consecutive rows share (g,r): in-lane logsumexp
            float l0 = lse4(acc[0], acc[1], acc[2], acc[3]);
            float l1 = lse4(acc[4], acc[5], acc[6], acc[7]);
            int group0 = m_tile * 4 + 2 * half;
            int col = (nt0 + t) * 16 + mn;
            float wm = -INFINITY;
            if (col < NCOL) {
                logits[(size_t)group0 * NCOL + col] = l0;
                logits[(size_t)(group0 + 1) * NCOL + col] = l1;
                wm = fmaxf(l0, l1);
            }
#pragma unroll
            for (int off = 16; off > 0; off >>= 1)
                wm = fmaxf(wm, __shfl_xor(wm, off, 32));
            if (lane == 0) atomicMax(&maxkey[m_tile / 5], fkey(wm));  // 5 m-tiles per graph
        }
    }
}

// ================= kernel 5: subtract per-graph max =================
__global__ void sub_max_kernel(float* __restrict__ logits, const int* __restrict__ maxkey) {
    size_t idx = (size_t)blockIdx.x * blockDim.x + threadIdx.x;
    if (idx >= (size_t)TOT_LOGITS) return;
    int g = (int)(idx / LOGITS_PER_G);
    logits[idx] -= funkey(maxkey[g]);
}

// ================= host-side launcher =================
extern "C" void kernel_launch(void* const* d_in, const int* in_sizes, int n_in,
                              void* d_out, int out_size, void* d_ws, size_t ws_size,
                              hipStream_t stream) {
    const float* focus   = (const float*)d_in[0];  // [32, 4608]
    const float* embt    = (const float*)d_in[1];  // [5, 768]
    const float* W       = (const float*)d_in[2];  // [6, 128, 80]
    const int*   species = (const int*)d_in[3];    // [32]
    float* out = (float*)d_out;                    // 92160 coeffs + 41,356,800 logits

    // workspace layout (~9.8 MB)
    char* ws = (char*)d_ws;
    double* nodes = (double*)ws;                       // 1440 B
    int*    maxk  = (int*)(ws + 2048);                 // 128 B
    float*  Amat  = (float*)(ws + 4096);               // 2560*36*4 = 368640 B
    float*  Yt    = (float*)(ws + 4096 + 368640);      // 36*65536*4 = 9,437,184 B

    float* out_coeffs = out;
    float* out_logits = out + COEFF_OUT;

    // 1) Gauss-Legendre nodes + max init
    gl_nodes_kernel<<<1, 192, 0, stream>>>(nodes, maxk);
    // 2) spherical-harmonics grid matrix Yt[36][NPAD2]
    ygrid_kernel<<<NPAD2 / 256, 256, 0, stream>>>(nodes, Yt);
    // 3) position_coeffs + reordered A matrix
    coeffs_kernel<<<NG * (LMAX + 1), 128, 0, stream>>>(focus, embt, W, species,
                                                       out_coeffs, Amat);
    // 4) blocked WMMA GEMM + fused logsumexp + per-graph max
    gemm_lse_kernel<<<MGROUPS * NCHUNKS, 256, 0, stream>>>(Amat, Yt, out_logits, maxk);
    // 5) max-normalize
    sub_max_kernel<<<(TOT_LOGITS + 255) / 256, 256, 0, stream>>>(out_logits, maxk);
}